// GATLayer_51788715655853
// MI455X (gfx1250) — compile-verified
//
#include <hip/hip_runtime.h>
#include <hip/hip_bf16.h>
#include <math.h>

// ---------------- problem constants (match reference) ----------------
static constexpr int NN = 100000;   // nodes
static constexpr int EE = 1600000;  // edges
static constexpr int HH = 4;        // heads
static constexpr int DD = 32;       // per-head dim
static constexpr int HD = HH * DD;  // 128
static constexpr float NEG_SLOPE = 0.2f;

typedef float v2f __attribute__((ext_vector_type(2)));
typedef float v8f __attribute__((ext_vector_type(8)));

// orderable-int mapping for float atomic max (involution)
__device__ __forceinline__ int f2key(float f) {
    int b = __float_as_int(f);
    return b ^ ((b >> 31) & 0x7fffffff);
}
__device__ __forceinline__ float key2f(int k) {
    return __int_as_float(k ^ ((k >> 31) & 0x7fffffff));
}

// ---------------- kernel 0: init accumulators ----------------
__global__ void gat_init_kernel(float* __restrict__ rst,
                                float* __restrict__ denom,
                                int* __restrict__ emax) {
    int idx = blockIdx.x * blockDim.x + threadIdx.x;
    if (idx < NN * HD) rst[idx] = 0.0f;
    if (idx < NN * HH) {
        denom[idx] = 0.0f;
        emax[idx]  = 0x807FFFFF;  // f2key(-inf)
    }
}

// ---------------- kernel 1: feat = h @ W_fc  (WMMA f32 16x16x4) ----------------
// grid = 6250 blocks (16 rows each), 256 threads = 8 waves; wave w -> col tile w (16 cols)
__global__ void gat_gemm_feat_kernel(const float* __restrict__ h,
                                     const float* __restrict__ W_fc,
                                     float* __restrict__ feat) {
    __shared__ float As[16 * 32];
    const int rowBase = blockIdx.x * 16;
    const int tid = threadIdx.x;
    for (int i = tid; i < 16 * 32; i += 256) As[i] = h[rowBase * 32 + i];
    __syncthreads();

    const int wave = tid >> 5;
    const int lane = tid & 31;
    const int colBase = wave * 16;
    const int r  = lane & 15;   // row within tile (A) / col within tile (B,C)
    const int hi = lane >> 4;   // half-wave selector

    v8f c = {};
#pragma unroll
    for (int kk = 0; kk < 8; ++kk) {
        const int kb = kk * 4;
        v2f a, b;
        // A 16x4 f32 layout: lanes0-15 hold K=kb,kb+1; lanes16-31 hold K=kb+2,kb+3
        a.x = As[r * 32 + kb + 2 * hi];
        a.y = As[r * 32 + kb + 2 * hi + 1];
        // B 4x16 f32 layout (row striped across lanes, symmetric halves)
        b.x = W_fc[(kb + 2 * hi) * HD + colBase + r];
        b.y = W_fc[(kb + 2 * hi + 1) * HD + colBase + r];
        c = __builtin_amdgcn_wmma_f32_16x16x4_f32(false, a, false, b,
                                                  (short)0, c, false, false);
    }
    // C/D layout: VGPR i -> row (i + 8*hi), col r
#pragma unroll
    for (int i = 0; i < 8; ++i)
        feat[(size_t)(rowBase + i + 8 * hi) * HD + colBase + r] = c[i];
}

// ---------------- kernel 2: el/er per (node, head) ----------------
__global__ void gat_elr_kernel(const float* __restrict__ feat,
                               const float* __restrict__ attn_l,
                               const float* __restrict__ attn_r,
                               float* __restrict__ el,
                               float* __restrict__ er) {
    int idx = blockIdx.x * blockDim.x + threadIdx.x;
    if (idx >= NN * HH) return;
    int n = idx >> 2, hh = idx & 3;
    const float4* f4 = (const float4*)(feat + (size_t)n * HD + hh * DD);
    const float4* al = (const float4*)(attn_l + hh * DD);
    const float4* ar = (const float4*)(attn_r + hh * DD);
    float sl = 0.f, sr = 0.f;
#pragma unroll
    for (int i = 0; i < 8; ++i) {
        float4 f = f4[i], a = al[i], b = ar[i];
        sl += f.x * a.x + f.y * a.y + f.z * a.z + f.w * a.w;
        sr += f.x * b.x + f.y * b.y + f.z * b.z + f.w * b.w;
    }
    el[idx] = sl;
    er[idx] = sr;
}

// ---------------- kernel 3: edge score + leaky relu + segment max ----------------
__global__ void gat_edge_score_kernel(const int* __restrict__ src,
                                      const int* __restrict__ dst,
                                      const float* __restrict__ el,
                                      const float* __restrict__ er,
                                      float* __restrict__ ebuf,
                                      int* __restrict__ emax) {
    int e = blockIdx.x * blockDim.x + threadIdx.x;
    if (e >= EE) return;
    int s = src[e], d = dst[e];
#pragma unroll
    for (int hh = 0; hh < HH; ++hh) {
        float v = el[s * HH + hh] + er[d * HH + hh];
        v = v > 0.f ? v : NEG_SLOPE * v;
        ebuf[(size_t)e * HH + hh] = v;
        atomicMax(&emax[d * HH + hh], f2key(v));
    }
}

// ---------------- kernel 4: edge exp + segment sum ----------------
__global__ void gat_edge_exp_kernel(const int* __restrict__ dst,
                                    float* __restrict__ ebuf,
                                    const int* __restrict__ emax,
                                    float* __restrict__ denom) {
    int e = blockIdx.x * blockDim.x + threadIdx.x;
    if (e >= EE) return;
    int d = dst[e];
#pragma unroll
    for (int hh = 0; hh < HH; ++hh) {
        float m = key2f(emax[d * HH + hh]);
        float x = __expf(ebuf[(size_t)e * HH + hh] - m);
        ebuf[(size_t)e * HH + hh] = x;
        atomicAdd(&denom[d * HH + hh], x);
    }
}

// ---------------- kernel 5: attention-weighted scatter-add ----------------
// one wave32 per edge; lane i handles feature dims [4i..4i+3] (head = i/8)
__global__ void gat_agg_kernel(const int* __restrict__ src,
                               const int* __restrict__ dst,
                               const float* __restrict__ feat,
                               const float* __restrict__ ebuf,
                               const float* __restrict__ denom,
                               float* __restrict__ rst) {
    int gid = blockIdx.x * blockDim.x + threadIdx.x;
    int e = gid >> 5;
    if (e >= EE) return;
    int lane = gid & 31;
    int s = src[e], d = dst[e];
    int hh = lane >> 3;
    float dn = denom[d * HH + hh];
    dn = dn > 1e-9f ? dn : 1e-9f;
    float alpha = ebuf[(size_t)e * HH + hh] / dn;
    float4 f = ((const float4*)feat)[(size_t)s * 32 + lane];
    float* out = rst + (size_t)d * HD + lane * 4;
    atomicAdd(out + 0, f.x * alpha);
    atomicAdd(out + 1, f.y * alpha);
    atomicAdd(out + 2, f.z * alpha);
    atomicAdd(out + 3, f.w * alpha);
}

// ---------------- kernel 6: out = ELU((rst+bias_gat)@W_out + b_out) + h ----------------
// grid = 6250 blocks, 64 threads = 2 waves; wave w -> col tile w; K = 128 in 32 WMMA steps
__global__ void gat_gemm_out_kernel(const float* __restrict__ rst,
                                    const float* __restrict__ bias_gat,
                                    const float* __restrict__ W_out,
                                    const float* __restrict__ b_out,
                                    const float* __restrict__ h,
                                    float* __restrict__ out) {
    __shared__ float As[16 * HD];
    const int rowBase = blockIdx.x * 16;
    const int tid = threadIdx.x;
    for (int i = tid; i < 16 * HD; i += 64)
        As[i] = rst[(size_t)rowBase * HD + i] + bias_gat[i & (HD - 1)];
    __syncthreads();

    const int wave = tid >> 5;
    const int lane = tid & 31;
    const int colBase = wave * 16;
    const int r  = lane & 15;
    const int hi = lane >> 4;

    v8f c = {};
#pragma unroll
    for (int kk = 0; kk < 32; ++kk) {
        const int kb = kk * 4;
        v2f a, b;
        a.x = As[r * HD + kb + 2 * hi];
        a.y = As[r * HD + kb + 2 * hi + 1];
        b.x = W_out[(kb + 2 * hi) * DD + colBase + r];
        b.y = W_out[(kb + 2 * hi + 1) * DD + colBase + r];
        c = __builtin_amdgcn_wmma_f32_16x16x4_f32(false, a, false, b,
                                                  (short)0, c, false, false);
    }
#pragma unroll
    for (int i = 0; i < 8; ++i) {
        int rg = rowBase + i + 8 * hi;
        int cg = colBase + r;
        float v = c[i] + b_out[cg];
        v = v > 0.f ? v : expm1f(v);   // ELU(alpha=1)
        out[(size_t)rg * DD + cg] = h[(size_t)rg * DD + cg] + v;
    }
}

// ---------------- launcher ----------------
extern "C" void kernel_launch(void* const* d_in, const int* in_sizes, int n_in,
                              void* d_out, int out_size, void* d_ws, size_t ws_size,
                              hipStream_t stream) {
    (void)in_sizes; (void)n_in; (void)out_size; (void)ws_size;

    const float* h        = (const float*)d_in[0];
    const int*   src      = (const int*)  d_in[1];
    const int*   dst      = (const int*)  d_in[2];
    const float* W_fc     = (const float*)d_in[3];
    const float* attn_l   = (const float*)d_in[4];
    const float* attn_r   = (const float*)d_in[5];
    const float* bias_gat = (const float*)d_in[6];
    const float* W_out    = (const float*)d_in[7];
    const float* b_out    = (const float*)d_in[8];
    float*       out      = (float*)d_out;

    // workspace carve-out (~134.4 MB of f32)
    float* feat  = (float*)d_ws;                       // N*128
    float* rst   = feat  + (size_t)NN * HD;            // N*128
    float* ebuf  = rst   + (size_t)NN * HD;            // E*4
    float* el    = ebuf  + (size_t)EE * HH;            // N*4
    float* er    = el    + (size_t)NN * HH;            // N*4
    float* denom = er    + (size_t)NN * HH;            // N*4
    int*   emax  = (int*)(denom + (size_t)NN * HH);    // N*4

    const int rowTiles = NN / 16;  // 6250 exactly

    // 0) init accumulators
    {
        int total = NN * HD;
        gat_init_kernel<<<(total + 255) / 256, 256, 0, stream>>>(rst, denom, emax);
    }
    // 1) feat = h @ W_fc
    gat_gemm_feat_kernel<<<rowTiles, 256, 0, stream>>>(h, W_fc, feat);
    // 2) el / er
    {
        int total = NN * HH;
        gat_elr_kernel<<<(total + 255) / 256, 256, 0, stream>>>(feat, attn_l, attn_r, el, er);
    }
    // 3) edge scores + segment max
    gat_edge_score_kernel<<<(EE + 255) / 256, 256, 0, stream>>>(src, dst, el, er, ebuf, emax);
    // 4) exp + segment sum
    gat_edge_exp_kernel<<<(EE + 255) / 256, 256, 0, stream>>>(dst, ebuf, emax, denom);
    // 5) weighted scatter-add (one wave per edge)
    {
        long long threads = (long long)EE * 32;
        int blocks = (int)((threads + 255) / 256);
        gat_agg_kernel<<<blocks, 256, 0, stream>>>(src, dst, feat, ebuf, denom, rst);
    }
    // 6) out projection + ELU + residual
    gat_gemm_out_kernel<<<rowTiles, 64, 0, stream>>>(rst, bias_gat, W_out, b_out, h, out);
}